// PlateYoloBlock_73916387164669
// MI455X (gfx1250) — compile-verified
//
#include <hip/hip_runtime.h>

// ---------------------------------------------------------------------------
// YOLO plate-decode: (32,13,256,256) NCHW f32 -> (32,65536,13) f32
// Memory-bound streaming kernel (208MB total @ 23.3TB/s ~= 9us floor).
// Path: NT global_load_b128 -> VALU transform -> ds_store_b128 staging ->
//       TDM tensor_store_from_lds (contiguous 53KB tile per workgroup).
// ---------------------------------------------------------------------------

typedef float        f32x4 __attribute__((ext_vector_type(4)));
typedef unsigned int u32x4 __attribute__((ext_vector_type(4)));
typedef int          i32x4 __attribute__((ext_vector_type(4)));
typedef int          i32x8 __attribute__((ext_vector_type(8)));

constexpr int C   = 13;                 // bbox attrs
constexpr int HW  = 256 * 256;          // pixels per batch image
constexpr int B   = 32;                 // batch
constexpr int PIX_PER_BLOCK = 1024;     // divides HW -> block never crosses batch
constexpr int THREADS       = 256;      // 4 pixels / thread, 8 waves
constexpr int NBLOCKS       = (B * HW) / PIX_PER_BLOCK;  // 2048

__device__ __forceinline__ float fast_sigmoid(float x) {
  float e = __expf(-x);                 // v_exp_f32 path
#if __has_builtin(__builtin_amdgcn_rcpf)
  return __builtin_amdgcn_rcpf(1.0f + e);
#else
  return 1.0f / (1.0f + e);
#endif
}

__global__ __launch_bounds__(THREADS)
void plate_yolo_decode(const float* __restrict__ in, float* __restrict__ out) {
  // Output staging tile: 1024 pixels * 13 attrs * 4B = 53,248 bytes
  __shared__ float tile[PIX_PER_BLOCK * C];

  const int t   = threadIdx.x;
  const int p0  = blockIdx.x * PIX_PER_BLOCK;   // first global pixel of block
  const int b   = p0 >> 16;                     // batch index (p0 / 65536)
  const int hw0 = (p0 & (HW - 1)) + t * 4;      // this thread's first in-plane pixel
  const int h   = hw0 >> 8;                     // row   (same for the 4 pixels)
  const int w0  = hw0 & 255;                    // col of first pixel (<= 252)

  const float* inb = in + ((size_t)(b * C) << 16);

  // ---- 13 coalesced non-temporal 128b loads (4 consecutive pixels each) ----
  f32x4 v[C];
#pragma unroll
  for (int c = 0; c < C; ++c) {
    v[c] = __builtin_nontemporal_load(
        reinterpret_cast<const f32x4*>(inb + ((size_t)c << 16) + hw0));
  }

  // ---- transform into the pixel-major output stream (52 floats) ----
  float vals[4 * C];
#pragma unroll
  for (int i = 0; i < 4; ++i) {
    vals[i * C + 0]  = (fast_sigmoid(v[0][i]) + (float)(w0 + i)) * 8.0f; // x
    vals[i * C + 1]  = (fast_sigmoid(v[1][i]) + (float)h) * 8.0f;        // y
    vals[i * C + 2]  = __expf(v[2][i]) * 8.0f;                           // w
    vals[i * C + 3]  = __expf(v[3][i]) * 8.0f;                           // h
#pragma unroll
    for (int c = 4; c < 12; ++c)
      vals[i * C + c] = v[c][i] * 8.0f;                                  // corners
    vals[i * C + 12] = fast_sigmoid(v[12][i]);                           // conf
  }

  // ---- 13 aligned b128 LDS stores: thread base = 208*t bytes (16B aligned) ----
  f32x4* dst = reinterpret_cast<f32x4*>(&tile[t * 4 * C]);
#pragma unroll
  for (int j = 0; j < C; ++j) {
    f32x4 q = { vals[4 * j + 0], vals[4 * j + 1],
                vals[4 * j + 2], vals[4 * j + 3] };
    dst[j] = q;
  }

  __syncthreads();   // all ds stores visible before the DMA reads LDS

  // ---- one wave issues the TDM store of the contiguous 53,248B tile ----
  if (t == 0) {
    const unsigned long long ga =
        (unsigned long long)(uintptr_t)(out + (size_t)p0 * C);
    // generic LDS address truncated to 32b == LDS byte offset (ISA 10.2)
    const unsigned int lds_addr = (unsigned int)(uintptr_t)(void*)&tile[0];
    const unsigned int N = PIX_PER_BLOCK * C;   // 13312 elements of 4B

    u32x4 g0;
    g0[0] = 1u;                                   // count=1, user descriptor
    g0[1] = lds_addr;                             // lds_addr [63:32]
    g0[2] = (unsigned int)ga;                     // global_addr[31:0]
    g0[3] = (unsigned int)((ga >> 32) & 0x01FFFFFFu) | 0x80000000u; // type=2

    i32x8 g1;
    g1[0] = 0x00020000;                           // data_size=2 (4 bytes)
    g1[1] = (int)((N & 0xFFFFu) << 16);           // tensor_dim0 lo (atomic addr=0)
    g1[2] = (int)((N >> 16) | (1u << 16));        // tensor_dim0 hi | tensor_dim1=1
    g1[3] = (int)((N & 0xFFFFu) << 16);           // tile_dim0 = 13312 (1-D tile)
    g1[4] = 0;                                    // tile_dim1=0, tile_dim2=0
    g1[5] = (int)N;                               // tensor_dim0_stride lo
    g1[6] = 0;                                    // stride hi | dim1_stride lo
    g1[7] = 0;                                    // dim1_stride hi
    i32x4 zz4 = {0, 0, 0, 0};                     // groups 2/3 unused (<=2D)
    i32x8 zz8 = {0, 0, 0, 0, 0, 0, 0, 0};         // extra arg (clang-23 6-arg form)

    __builtin_amdgcn_tensor_store_from_lds(g0, g1, zz4, zz4, zz8, 0);
    __builtin_amdgcn_s_wait_tensorcnt(0);         // explicit; s_endpgm also waits
  }
}

extern "C" void kernel_launch(void* const* d_in, const int* in_sizes, int n_in,
                              void* d_out, int out_size, void* d_ws, size_t ws_size,
                              hipStream_t stream) {
  const float* coords = (const float*)d_in[0];   // (32,13,256,256) f32
  float* out = (float*)d_out;                    // (32,65536,13) f32
  plate_yolo_decode<<<NBLOCKS, THREADS, 0, stream>>>(coords, out);
}